// IF_24627342475908
// MI455X (gfx1250) — compile-verified
//
#include <hip/hip_runtime.h>

// IF spiking neuron forward (ZIF Heaviside), T-step scan, memory-bound.
// T=8, B=32, C=128, H=32, W=32.  x:[T*B,C,H,W] f32, thresh:[1] f32, bias:[T,C] f32.
// out[t,b,c,h,w] = spike stream. Floor: 268 MB @ 23.3 TB/s ~= 11.5 us.

#define TT 8
#define BB 32
#define CC 128
#define HH 32
#define WW 32
#define PLANE (HH * WW)              /* 1024 floats per (b,c) plane        */
#define STRIDE_T (BB * CC * PLANE)   /* 4,194,304 floats between timesteps */

typedef __attribute__((ext_vector_type(4))) float v4f;

__global__ __launch_bounds__(256) void if_neuron_kernel(
    const float* __restrict__ x,
    const float* __restrict__ thresh,
    const float* __restrict__ bias,
    float* __restrict__ out)
{
    // One (b,c) plane per block: 256 lanes x float4 = 1024 floats = H*W.
    __shared__ v4f smem[TT * 256];   // 32 KB -> 10 blocks/WGP fit in 320 KB LDS

    const int tid   = threadIdx.x;
    const int plane = blockIdx.x;              // = b*C + c
    const int c     = plane & (CC - 1);        // uniform per block
    const int base  = plane * PLANE + tid * 4; // f32 element offset within one timestep

    // Kick off all 8 async HBM->LDS b128 copies (ASYNCcnt path). Each lane
    // moves its own 16B chunk for each timestep; the wave is free to do the
    // scalar setup while the async engine streams from HBM.
    #pragma unroll
    for (int t = 0; t < TT; ++t) {
        const float* gptr = x + (size_t)t * STRIDE_T + base;
        // Low 32 bits of a flat LDS pointer are the allocation-relative LDS
        // byte offset (aperture rules) -- exactly what the async op's VDST wants.
        unsigned lds_off = (unsigned)(uintptr_t)(&smem[t * 256 + tid]);
        asm volatile("global_load_async_to_lds_b128 %0, %1, off"
                     :: "v"(lds_off), "v"(gptr)
                     : "memory");
    }

    const float thre = thresh[0];              // uniform -> s_load
    v4f mem = { 0.5f * thre, 0.5f * thre, 0.5f * thre, 0.5f * thre };

    // Wait for all async transfers. Each lane reads back only the LDS slots
    // it targeted itself, and ASYNCcnt is a per-wave counter covering those
    // lanes' transfers -> no workgroup barrier required.
    asm volatile("s_wait_asynccnt 0" ::: "memory");

    #pragma unroll
    for (int t = 0; t < TT; ++t) {
        const float bt = bias[t * CC + c];     // block-uniform -> scalar load
        v4f xt = smem[t * 256 + tid];          // ds_load_b128

        mem = mem + xt - bt;

        v4f sp;
        sp.x = (mem.x >= thre) ? thre : 0.0f;  // zif(mem - thresh) * thresh
        sp.y = (mem.y >= thre) ? thre : 0.0f;
        sp.z = (mem.z >= thre) ? thre : 0.0f;
        sp.w = (mem.w >= thre) ? thre : 0.0f;
        mem = mem - sp;

        // NT store: 128 MiB pure output stream; keep it from thrashing the
        // 192 MB L2 that is caching the x stream.
        __builtin_nontemporal_store(
            sp, reinterpret_cast<v4f*>(out + (size_t)t * STRIDE_T + base));
    }
}

extern "C" void kernel_launch(void* const* d_in, const int* in_sizes, int n_in,
                              void* d_out, int out_size, void* d_ws, size_t ws_size,
                              hipStream_t stream) {
    (void)in_sizes; (void)n_in; (void)out_size; (void)d_ws; (void)ws_size;
    const float* x      = (const float*)d_in[0];
    const float* thresh = (const float*)d_in[1];
    const float* bias   = (const float*)d_in[2];
    float*       out    = (float*)d_out;

    if_neuron_kernel<<<dim3(BB * CC), dim3(256), 0, stream>>>(x, thresh, bias, out);
}